// BipartiteLoss_10892037063325
// MI455X (gfx1250) — compile-verified
//
#include <hip/hip_runtime.h>

#define B_    1024
#define N_    100      // queries
#define M_    20       // targets
#define C_    92       // classes
#define NPAD  112      // N padded to 7*16 for WMMA row tiles
#define INF_  1e30f

typedef float v2f __attribute__((ext_vector_type(2)));
typedef float v8f __attribute__((ext_vector_type(8)));

__device__ __forceinline__ float blockReduceSum128(float v, float* red, int tid) {
    red[tid] = v; __syncthreads();
    for (int s = 64; s > 0; s >>= 1) {
        if (tid < s) red[tid] += red[tid + s];
        __syncthreads();
    }
    float r = red[0]; __syncthreads();
    return r;
}

__global__ __launch_bounds__(128) void detr_batch_kernel(
    const float* __restrict__ cls_pred,   // [B, N, C]
    const float* __restrict__ bb_pred,    // [B, N, 4] cxcywh
    const int*   __restrict__ cls_gt,     // [B, M]
    const float* __restrict__ bb_gt,      // [B, M, 4] cxcywh
    float* __restrict__ partials)         // [B, 3] (ce_mean, bbox_mean, giou_mean)
{
    __shared__ float probs[NPAD][C_];     // logits -> softmax probs (rows 100..111 zero)
    __shared__ float lse[N_];
    __shared__ float costT[M_][NPAD];     // cost^T : rows=targets, cols=queries
    __shared__ float praw[N_][4], graw[M_][4];        // raw cxcywh
    __shared__ float px[N_][4], gx[M_][4];            // xyxy
    __shared__ float pa[N_], ga[M_];                  // areas
    __shared__ int   sgt[32];                         // gt classes, -1 padded
    __shared__ float u_[M_ + 1], v_[N_ + 1], minv[N_ + 1];
    __shared__ int   way[N_ + 1], p_[N_ + 1], used[N_ + 1];
    __shared__ int   pidx[M_], tidx[M_], tc[N_];
    __shared__ float red[128], red2[128];
    __shared__ int   redi[128];
    __shared__ int   j0sh, i0sh;

    const int tid = threadIdx.x;
    const int b   = blockIdx.x;
    const size_t clsBase = (size_t)b * N_ * C_;
    const size_t bbBase  = (size_t)b * N_ * 4;
    const size_t gtBase  = (size_t)b * M_;
    const size_t gbBase  = (size_t)b * M_ * 4;

    // ---- Load logits (coalesced) + zero pad rows; load boxes & gt classes ----
    for (int i = tid; i < N_ * C_; i += 128)
        probs[i / C_][i % C_] = cls_pred[clsBase + i];
    for (int i = N_ * C_ + tid; i < NPAD * C_; i += 128)
        probs[i / C_][i % C_] = 0.0f;
    if (tid < 32) sgt[tid] = (tid < M_) ? cls_gt[gtBase + tid] : -1;
    if (tid < N_) {
        float cx = bb_pred[bbBase + tid*4 + 0], cy = bb_pred[bbBase + tid*4 + 1];
        float w  = bb_pred[bbBase + tid*4 + 2], h  = bb_pred[bbBase + tid*4 + 3];
        praw[tid][0]=cx; praw[tid][1]=cy; praw[tid][2]=w; praw[tid][3]=h;
        float x0 = cx - 0.5f*w, y0 = cy - 0.5f*h, x1 = cx + 0.5f*w, y1 = cy + 0.5f*h;
        px[tid][0]=x0; px[tid][1]=y0; px[tid][2]=x1; px[tid][3]=y1;
        pa[tid] = (x1 - x0) * (y1 - y0);
    }
    if (tid < M_) {
        float cx = bb_gt[gbBase + tid*4 + 0], cy = bb_gt[gbBase + tid*4 + 1];
        float w  = bb_gt[gbBase + tid*4 + 2], h  = bb_gt[gbBase + tid*4 + 3];
        graw[tid][0]=cx; graw[tid][1]=cy; graw[tid][2]=w; graw[tid][3]=h;
        float x0 = cx - 0.5f*w, y0 = cy - 0.5f*h, x1 = cx + 0.5f*w, y1 = cy + 0.5f*h;
        gx[tid][0]=x0; gx[tid][1]=y0; gx[tid][2]=x1; gx[tid][3]=y1;
        ga[tid] = (x1 - x0) * (y1 - y0);
    }
    __syncthreads();

    // ---- Softmax per query row (in place) ----
    if (tid < N_) {
        float mx = -INF_;
        for (int c = 0; c < C_; ++c) mx = fmaxf(mx, probs[tid][c]);
        float s = 0.0f;
        for (int c = 0; c < C_; ++c) { float e = expf(probs[tid][c] - mx); probs[tid][c] = e; s += e; }
        float inv = 1.0f / s;
        for (int c = 0; c < C_; ++c) probs[tid][c] *= inv;
        lse[tid] = mx + logf(s);
    }
    __syncthreads();

    // ---- cost bbox + iou terms: costT[t][q] = 5*L1 - 2*IoU ----
    for (int idx = tid; idx < M_ * N_; idx += 128) {
        int t = idx / N_, q = idx % N_;
        float l1 = fabsf(praw[q][0]-graw[t][0]) + fabsf(praw[q][1]-graw[t][1])
                 + fabsf(praw[q][2]-graw[t][2]) + fabsf(praw[q][3]-graw[t][3]);
        float x0 = fmaxf(px[q][0], gx[t][0]), y0 = fmaxf(px[q][1], gx[t][1]);
        float x1 = fminf(px[q][2], gx[t][2]), y1 = fminf(px[q][3], gx[t][3]);
        float iw = fmaxf(x1 - x0, 0.0f), ih = fmaxf(y1 - y0, 0.0f);
        float inter = iw * ih;
        float iou = inter / (pa[q] + ga[t] - inter);
        costT[t][q] = 5.0f * l1 - 2.0f * iou;
    }
    __syncthreads();

    // ---- WMMA f32: cost_class = -(P[100x92] @ onehot(gt)[92x20]) ----
    {
        const int wave = tid >> 5, lane = tid & 31;
        const int lh   = (lane >= 16) ? 1 : 0;   // which K-half this lane holds
        const int l15  = lane & 15;
        for (int tile = wave; tile < 14; tile += 4) {   // 7 row-tiles x 2 col-tiles
            int rt = tile >> 1, mt = tile & 1;
            int row = rt * 16 + l15;                    // A row (query), zero-padded rows ok
            int col = mt * 16 + l15;                    // B/D col (target), sgt=-1 padded
            int g = sgt[col];
            v8f acc = {0.f,0.f,0.f,0.f,0.f,0.f,0.f,0.f};
            for (int kt = 0; kt < 23; ++kt) {           // K = 92 = 23*4
                int kb = kt * 4 + lh * 2;
                v2f a;  a.x  = probs[row][kb];            a.y  = probs[row][kb + 1];
                v2f bm; bm.x = (g == kb)     ? 1.0f : 0.0f;
                        bm.y = (g == kb + 1) ? 1.0f : 0.0f;
                acc = __builtin_amdgcn_wmma_f32_16x16x4_f32(
                        false, a, false, bm, (short)0, acc, false, false);
            }
            // D: vgpr j -> row j + 8*lh of tile, col = l15
            for (int j = 0; j < 8; ++j) {
                int q = rt * 16 + j + 8 * lh;
                if (q < N_ && col < M_) costT[col][q] -= acc[j];
            }
        }
    }
    __syncthreads();

    // ---- Hungarian (JV) on costT [20 x 100], column scan parallelized ----
    if (tid <= M_) u_[tid] = 0.0f;
    if (tid <= N_) { v_[tid] = 0.0f; p_[tid] = 0; }
    __syncthreads();

    for (int i = 1; i <= M_; ++i) {
        if (tid == 0) { p_[0] = i; j0sh = 0; }
        if (tid <= N_) { minv[tid] = INF_; used[tid] = 0; }
        __syncthreads();
        while (true) {
            if (tid == 0) { used[j0sh] = 1; i0sh = p_[j0sh]; }
            __syncthreads();
            int j0 = j0sh, i0 = i0sh;
            float ui0 = u_[i0];
            if (tid < N_) {
                int j = tid + 1;
                if (!used[j]) {
                    float cur = costT[i0 - 1][j - 1] - ui0 - v_[j];
                    if (cur < minv[j]) { minv[j] = cur; way[j] = j0; }
                }
            }
            __syncthreads();
            // argmin over free columns (first-min tie-break like np.argmin)
            float val = INF_; int vi = 0x7fffffff;
            if (tid < N_ && !used[tid + 1]) { val = minv[tid + 1]; vi = tid + 1; }
            red[tid] = val; redi[tid] = vi;
            __syncthreads();
            for (int s = 64; s > 0; s >>= 1) {
                if (tid < s) {
                    float ov = red[tid + s]; int oi = redi[tid + s];
                    if (ov < red[tid] || (ov == red[tid] && oi < redi[tid])) {
                        red[tid] = ov; redi[tid] = oi;
                    }
                }
                __syncthreads();
            }
            float delta = red[0]; int j1 = redi[0];
            if (tid <= N_) {
                if (used[tid]) { u_[p_[tid]] += delta; v_[tid] -= delta; }
                else if (tid >= 1) minv[tid] -= delta;
            }
            if (tid == 0) j0sh = j1;
            __syncthreads();
            if (p_[j1] == 0) break;
        }
        if (tid == 0) {               // augmenting path
            int j0 = j0sh;
            while (j0) { int jj = way[j0]; p_[j0] = p_[jj]; j0 = jj; }
        }
        __syncthreads();
    }

    // ---- Extract assignment (cols ascending, like scipy/nonzero order) ----
    if (tid == 0) {
        int k = 0;
        for (int j = 1; j <= N_; ++j)
            if (p_[j] != 0) { pidx[k] = j - 1; tidx[k] = p_[j] - 1; ++k; }
    }
    if (tid < N_) tc[tid] = C_ - 1;
    __syncthreads();
    if (tid == 0)
        for (int k = 0; k < M_; ++k) tc[pidx[k]] = sgt[tidx[k]];
    __syncthreads();

    // ---- loss_ce (weighted mean per batch) ----
    float l_nw = 0.0f, l_w = 0.0f;
    if (tid < N_) {
        int c = tc[tid];
        float logit = cls_pred[clsBase + (size_t)tid * C_ + c];
        float nll = lse[tid] - logit;
        float wt  = (c == C_ - 1) ? 0.1f : 1.0f;
        l_nw = nll * wt; l_w = wt;
    }
    float s_nw = blockReduceSum128(l_nw, red, tid);
    float s_w  = blockReduceSum128(l_w,  red2, tid);

    // ---- loss_bbox (mean over M*4) ----
    float l_bb = 0.0f;
    if (tid < M_ * 4) {
        int k = tid >> 2, d = tid & 3;
        l_bb = fabsf(praw[pidx[k]][d] - graw[tidx[k]][d]);
    }
    float s_bb = blockReduceSum128(l_bb, red, tid);

    // ---- loss_giou: mean of (1 - iou) over the FULL M x M matrix ----
    float l_iou = 0.0f;
    for (int idx = tid; idx < M_ * M_; idx += 128) {
        int k = idx / M_, l = idx % M_;
        int q = pidx[k], t = tidx[l];
        float x0 = fmaxf(px[q][0], gx[t][0]), y0 = fmaxf(px[q][1], gx[t][1]);
        float x1 = fminf(px[q][2], gx[t][2]), y1 = fminf(px[q][3], gx[t][3]);
        float iw = fmaxf(x1 - x0, 0.0f), ih = fmaxf(y1 - y0, 0.0f);
        float inter = iw * ih;
        l_iou += inter / (pa[q] + ga[t] - inter);
    }
    float s_iou = blockReduceSum128(l_iou, red, tid);

    if (tid == 0) {
        partials[b * 3 + 0] = s_nw / s_w;
        partials[b * 3 + 1] = s_bb * (1.0f / (M_ * 4));
        partials[b * 3 + 2] = 1.0f - s_iou * (1.0f / (M_ * M_));
    }
}

__global__ __launch_bounds__(256) void detr_finalize_kernel(
    const float* __restrict__ partials, float* __restrict__ out)
{
    __shared__ float red[256];
    int tid = threadIdx.x;
    float sums[3];
    for (int c = 0; c < 3; ++c) {
        float s = 0.0f;
        for (int b = tid; b < B_; b += 256) s += partials[b * 3 + c];  // fixed order
        red[tid] = s; __syncthreads();
        for (int st = 128; st > 0; st >>= 1) {
            if (tid < st) red[tid] += red[tid + st];
            __syncthreads();
        }
        sums[c] = red[0]; __syncthreads();
    }
    if (tid == 0) {
        float ce = sums[0] * (1.0f / B_);
        float bb = sums[1] * (5.0f / B_);
        float gi = sums[2] * (2.0f / B_);
        out[0] = ce; out[1] = bb; out[2] = gi; out[3] = ce + bb + gi;
    }
}

extern "C" void kernel_launch(void* const* d_in, const int* in_sizes, int n_in,
                              void* d_out, int out_size, void* d_ws, size_t ws_size,
                              hipStream_t stream) {
    const float* cls_pred = (const float*)d_in[0];
    const float* bb_pred  = (const float*)d_in[1];
    const int*   cls_gt   = (const int*)d_in[2];
    const float* bb_gt    = (const float*)d_in[3];
    float* out      = (float*)d_out;
    float* partials = (float*)d_ws;   // B_*3 floats = 12 KB

    detr_batch_kernel<<<B_, 128, 0, stream>>>(cls_pred, bb_pred, cls_gt, bb_gt, partials);
    detr_finalize_kernel<<<1, 256, 0, stream>>>(partials, out);
}